// EnhancedGATCN_41549513621695
// MI455X (gfx1250) — compile-verified
//
#include <hip/hip_runtime.h>
#include <math.h>

// ---------------------------------------------------------------------------
// EnhancedGATCN on MI455X (gfx1250): f32 WMMA GEMMs + L2-resident edge phase.
// ---------------------------------------------------------------------------

typedef __attribute__((ext_vector_type(2))) float v2f;
typedef __attribute__((ext_vector_type(8))) float v8f;

#define IN_CH 128
#define EXT   3
#define HID   64
#define NCOL  80   // 64 features + alpha_src + alpha_dst + pad

__device__ inline void atomAddF(float* addr, float v) {
    __hip_atomic_fetch_add(addr, v, __ATOMIC_RELAXED, __HIP_MEMORY_SCOPE_AGENT);
}

// float atomic max via ordered-int trick (valid with -inf init, mixed signs)
__device__ inline void atomMaxF(float* addr, float val) {
    if (val >= 0.0f) atomicMax((int*)addr, __float_as_int(val));
    else             atomicMin((unsigned int*)addr, __float_as_uint(val));
}

// ---------------------------------------------------------------------------
// Fold attention vectors into the weight matrix:
//   Wc[k][0..63] = W[k][*], Wc[k][64] = W[k]·a_src, Wc[k][65] = W[k]·a_dst
// Rows k >= K are zero padding (for K=131 -> KPAD=132).
// ---------------------------------------------------------------------------
__global__ void build_wc(const float* __restrict__ W, const float* __restrict__ a_src,
                         const float* __restrict__ a_dst, float* __restrict__ Wc,
                         int K, int KPAD) {
    int k = blockIdx.x * blockDim.x + threadIdx.x;
    if (k >= KPAD) return;
    float s0 = 0.f, s1 = 0.f;
    for (int j = 0; j < HID; ++j) {
        float w = (k < K) ? W[k * HID + j] : 0.f;
        Wc[k * NCOL + j] = w;
        s0 += w * a_src[j];
        s1 += w * a_dst[j];
    }
    Wc[k * NCOL + 64] = s0;
    Wc[k * NCOL + 65] = s1;
    for (int j = 66; j < NCOL; ++j) Wc[k * NCOL + j] = 0.f;
}

// ce = We(row 0) · a_edge   (EDGE_DIM == 1)
__global__ void build_ce(const float* __restrict__ We, const float* __restrict__ ae,
                         float* __restrict__ ce) {
    if (threadIdx.x == 0 && blockIdx.x == 0) {
        float s = 0.f;
        for (int j = 0; j < HID; ++j) s += We[j] * ae[j];
        ce[0] = s;
    }
}

// ---------------------------------------------------------------------------
// WMMA GEMM: [N, KPAD] x [KPAD, 80] -> H[N,64], alpha_src[N], alpha_dst[N].
// Input A is the (optionally concatenated) node features: col < k0 from A0,
// next kext cols from Aext, rest zero.
// Block = 128 threads (4 waves); each wave owns a 16-row tile, loops over
// 5 column tiles of 16; inner K loop in steps of 4 via v_wmma_f32_16x16x4_f32.
// A fragments live in registers; B tiles come from LDS.
// WMMA f32 layouts (ISA 7.12.2):
//   A 16x4 : lane = (k>>1&1)*16 + m, component = k&1     (k relative to step)
//   B 4x16 : lane = (k>>1&1)*16 + n, component = k&1
//   C 16x16: vgpr r, lanes[16..31] -> m += 8;  m = r + (lane>>4)*8, n = lane&15
// ---------------------------------------------------------------------------
template <int KPAD>
__global__ __launch_bounds__(128) void gat_gemm_wmma(
    const float* __restrict__ A0, int k0,
    const float* __restrict__ Aext, int kext,
    const float* __restrict__ Wc,          // [KPAD][NCOL]
    float* __restrict__ Hout,              // [N][64]
    float* __restrict__ asrc, float* __restrict__ adst, int N) {
    __shared__ float sW[KPAD * NCOL];

    const int tid  = threadIdx.x;
    const int wave = tid >> 5;
    const int lane = tid & 31;
    const int row0 = blockIdx.x * 64;

    for (int i = tid; i < KPAD * NCOL; i += 128) sW[i] = Wc[i];
    __syncthreads();

    const int m  = lane & 15;            // row within 16-tile (also col for B)
    const int kh = (lane >> 4) * 2;      // K sub-offset from lane half
    const int gr = row0 + wave * 16 + m; // global row this lane's A-frag uses
    const bool rok = (gr < N);

    // Preload all A fragments for this wave's 16-row tile into registers.
    constexpr int NKT = KPAD / 4;
    v2f afr[NKT];
#pragma unroll
    for (int i = 0; i < NKT; ++i) {
        int c0 = i * 4 + kh;
        float x0 = 0.f, x1 = 0.f;
        if (rok) {
            int c = c0;
            if (c < k0)              x0 = A0[(long)gr * k0 + c];
            else if (c < k0 + kext)  x0 = Aext[(long)gr * kext + (c - k0)];
            c = c0 + 1;
            if (c < k0)              x1 = A0[(long)gr * k0 + c];
            else if (c < k0 + kext)  x1 = Aext[(long)gr * kext + (c - k0)];
        }
        afr[i].x = x0;
        afr[i].y = x1;
    }

    const int gm_base = row0 + wave * 16 + (lane >> 4) * 8;

    for (int ct = 0; ct < 5; ++ct) {
        const int n = ct * 16 + (lane & 15);
        v8f acc = {0.f, 0.f, 0.f, 0.f, 0.f, 0.f, 0.f, 0.f};
#pragma unroll
        for (int i = 0; i < NKT; ++i) {
            v2f b;
            b.x = sW[(i * 4 + kh) * NCOL + n];
            b.y = sW[(i * 4 + kh + 1) * NCOL + n];
            acc = __builtin_amdgcn_wmma_f32_16x16x4_f32(
                false, afr[i], false, b, (short)0, acc, false, false);
        }
#pragma unroll
        for (int r = 0; r < 8; ++r) {
            int gm = gm_base + r;
            if (gm < N) {
                if (ct < 4) {
                    Hout[(long)gm * HID + ct * 16 + (lane & 15)] = acc[r];
                } else {
                    int col = lane & 15;
                    if (col == 0)      asrc[gm] = acc[r];
                    else if (col == 1) adst[gm] = acc[r];
                }
            }
        }
    }
}

// ---------------------------------------------------------------------------
// Edge phase kernels
// ---------------------------------------------------------------------------
__global__ void init_seg(float* __restrict__ agg, float* __restrict__ maxv,
                         float* __restrict__ denom, int N) {
    int i = blockIdx.x * blockDim.x + threadIdx.x;
    if (i < N * HID) agg[i] = 0.f;
    if (i < N) { maxv[i] = -INFINITY; denom[i] = 0.f; }
}

__global__ void edge_logits_max(const int* __restrict__ src, const int* __restrict__ dst,
                                const float* __restrict__ ew, const float* __restrict__ ce,
                                const float* __restrict__ asrc, const float* __restrict__ adst,
                                float* __restrict__ logit, float* __restrict__ maxv, int E) {
    int e = blockIdx.x * blockDim.x + threadIdx.x;
    if (e >= E) return;
    int s = src[e], d = dst[e];
    float l = asrc[s] + adst[d] + ew[e] * ce[0];
    l = (l > 0.f) ? l : 0.2f * l;  // leaky_relu(0.2)
    logit[e] = l;
    atomMaxF(&maxv[d], l);
}

__global__ void edge_exp_sum(const int* __restrict__ dst, const float* __restrict__ maxv,
                             float* __restrict__ exv, float* __restrict__ denom, int E) {
    int e = blockIdx.x * blockDim.x + threadIdx.x;
    if (e >= E) return;
    int d = dst[e];
    float ex = __expf(exv[e] - maxv[d]);
    exv[e] = ex;
    atomAddF(&denom[d], ex);
}

// One wave32 per edge: lane covers features c and c+32 -> coalesced h[src] reads.
__global__ __launch_bounds__(256) void edge_scatter(
    const int* __restrict__ src, const int* __restrict__ dst,
    const float* __restrict__ exv, const float* __restrict__ denom,
    const float* __restrict__ h, float* __restrict__ agg, int E) {
    int e = blockIdx.x * 8 + (threadIdx.x >> 5);
    if (e >= E) return;
    int lane = threadIdx.x & 31;
    int s = src[e], d = dst[e];
    float coef = exv[e] / (denom[d] + 1e-16f);
    float v0 = h[(long)s * HID + lane]      * coef;
    float v1 = h[(long)s * HID + 32 + lane] * coef;
    atomAddF(&agg[(long)d * HID + lane],      v0);
    atomAddF(&agg[(long)d * HID + 32 + lane], v1);
}

__global__ void bias_relu(float* __restrict__ agg, const float* __restrict__ b, int total) {
    int i = blockIdx.x * blockDim.x + threadIdx.x;
    if (i >= total) return;
    float v = agg[i] + b[i & (HID - 1)];
    agg[i] = (v > 0.f) ? v : 0.f;
}

// out = relu(concat(h2, x_ext) @ Wlin + blin)  -> [N,2]
__global__ void final_linear(const float* __restrict__ h, const float* __restrict__ xext,
                             const float* __restrict__ Wlin, const float* __restrict__ blin,
                             float* __restrict__ out, int N) {
    int mrow = blockIdx.x * blockDim.x + threadIdx.x;
    if (mrow >= N) return;
    float a0 = blin[0], a1 = blin[1];
    const float* hr = h + (long)mrow * HID;
    for (int i = 0; i < HID; ++i) {
        float v = hr[i];
        a0 += v * Wlin[2 * i];
        a1 += v * Wlin[2 * i + 1];
    }
    for (int i = 0; i < EXT; ++i) {
        float v = xext[(long)mrow * EXT + i];
        a0 += v * Wlin[2 * (HID + i)];
        a1 += v * Wlin[2 * (HID + i) + 1];
    }
    out[2 * mrow]     = (a0 > 0.f) ? a0 : 0.f;
    out[2 * mrow + 1] = (a1 > 0.f) ? a1 : 0.f;
}

// ---------------------------------------------------------------------------
extern "C" void kernel_launch(void* const* d_in, const int* in_sizes, int n_in,
                              void* d_out, int out_size, void* d_ws, size_t ws_size,
                              hipStream_t stream) {
    const float* x      = (const float*)d_in[0];
    const float* xext   = (const float*)d_in[1];
    const int*   eidx   = (const int*)d_in[2];
    const float* ew     = (const float*)d_in[3];
    const float* W1     = (const float*)d_in[4];
    const float* as1    = (const float*)d_in[5];
    const float* ad1    = (const float*)d_in[6];
    const float* We1    = (const float*)d_in[7];
    const float* ae1    = (const float*)d_in[8];
    const float* b1     = (const float*)d_in[9];
    const float* W2     = (const float*)d_in[10];
    const float* as2    = (const float*)d_in[11];
    const float* ad2    = (const float*)d_in[12];
    const float* We2    = (const float*)d_in[13];
    const float* ae2    = (const float*)d_in[14];
    const float* b2     = (const float*)d_in[15];
    const float* Wlin   = (const float*)d_in[16];
    const float* blin   = (const float*)d_in[17];
    float* out = (float*)d_out;

    const int N = in_sizes[0] / IN_CH;
    const int E = in_sizes[2] / 2;
    const int* src = eidx;
    const int* dst = eidx + E;

    // workspace layout (floats)
    float* ws   = (float*)d_ws;
    const long N64 = (long)N * HID;
    float* bufA  = ws;             // [N*64]  gemm output h
    float* bufB  = bufA + N64;     // [N*64]  aggregation / next-layer input
    float* asrcv = bufB + N64;     // [N]
    float* adstv = asrcv + N;      // [N]
    float* maxv  = adstv + N;      // [N]
    float* denom = maxv + N;       // [N]
    float* exv   = denom + N;      // [E]
    float* wc    = exv + E;        // [132*80]
    float* ce    = wc + 132 * NCOL;// [1]

    const int nodeBlocks  = (N + 63) / 64;
    const int edgeBlocks  = (E + 255) / 256;
    const int scatBlocks  = (E + 7) / 8;       // 8 waves (edges) per 256-thr block
    const int segBlocks   = (int)((N64 + 255) / 256);

    // ---------------- layer 1 ----------------
    build_wc<<<(132 + 127) / 128, 128, 0, stream>>>(W1, as1, ad1, wc, 131, 132);
    build_ce<<<1, 64, 0, stream>>>(We1, ae1, ce);
    gat_gemm_wmma<132><<<nodeBlocks, 128, 0, stream>>>(x, IN_CH, xext, EXT, wc,
                                                       bufA, asrcv, adstv, N);
    init_seg<<<segBlocks, 256, 0, stream>>>(bufB, maxv, denom, N);
    edge_logits_max<<<edgeBlocks, 256, 0, stream>>>(src, dst, ew, ce, asrcv, adstv,
                                                    exv, maxv, E);
    edge_exp_sum<<<edgeBlocks, 256, 0, stream>>>(dst, maxv, exv, denom, E);
    edge_scatter<<<scatBlocks, 256, 0, stream>>>(src, dst, exv, denom, bufA, bufB, E);
    bias_relu<<<segBlocks, 256, 0, stream>>>(bufB, b1, (int)N64);

    // ---------------- layer 2 ----------------
    build_wc<<<1, 128, 0, stream>>>(W2, as2, ad2, wc, 64, 64);
    build_ce<<<1, 64, 0, stream>>>(We2, ae2, ce);
    gat_gemm_wmma<64><<<nodeBlocks, 128, 0, stream>>>(bufB, HID, nullptr, 0, wc,
                                                      bufA, asrcv, adstv, N);
    init_seg<<<segBlocks, 256, 0, stream>>>(bufB, maxv, denom, N);
    edge_logits_max<<<edgeBlocks, 256, 0, stream>>>(src, dst, ew, ce, asrcv, adstv,
                                                    exv, maxv, E);
    edge_exp_sum<<<edgeBlocks, 256, 0, stream>>>(dst, maxv, exv, denom, E);
    edge_scatter<<<scatBlocks, 256, 0, stream>>>(src, dst, exv, denom, bufA, bufB, E);
    bias_relu<<<segBlocks, 256, 0, stream>>>(bufB, b2, (int)N64);

    // ---------------- head ----------------
    final_linear<<<(N + 255) / 256, 256, 0, stream>>>(bufB, xext, Wlin, blin, out, N);
}